// FNSSelfAttention_41747082117448
// MI455X (gfx1250) — compile-verified
//
#include <hip/hip_runtime.h>
#include <math.h>

// ---------------------------------------------------------------------------
// FNS self-attention (fractional kernel attention) for MI455X / gfx1250.
// fp32 end-to-end; both GEMMs use V_WMMA_F32_16X16X4_F32 (wave32 WMMA).
// Bandwidth-optimized: causal tiles skipped, reductions fused into the score
// pass (deterministic tree reduction, no float atomics), b128 loads/stores,
// V transposed once (LDS-tiled) so both GEMM operands stream as b128.
// ---------------------------------------------------------------------------

typedef float v2f __attribute__((ext_vector_type(2)));
typedef float v8f __attribute__((ext_vector_type(8)));

namespace fns {
constexpr int B  = 2;
constexpr int H  = 12;
constexpr int N  = 2048;
constexpr int D  = 64;
constexpr int BH = B * H;
constexpr int NT = N / 16;                    // 128 tiles per dimension
constexpr float EXPONENT = -(64.0f + 1.5f);   // -(d_intrinsic + ALPHA)
constexpr float EPS      = 1e-12f;
}

#define WMMA_F32X4(a, b, c) \
  __builtin_amdgcn_wmma_f32_16x16x4_f32(false, (a), false, (b), (short)0, (c), false, false)

// ---------------------------------------------------------------------------
// 1) L2-normalize rows of length 64: one wave per row, float2 per lane.
// ---------------------------------------------------------------------------
__global__ void fns_l2norm(const float* __restrict__ x, float* __restrict__ y,
                           int nrows) {
  using namespace fns;
  const int wave = (int)((blockIdx.x * blockDim.x + threadIdx.x) >> 5);
  const int lane = threadIdx.x & 31;
  if (wave >= nrows) return;
  const float2 v = ((const float2*)(x + (size_t)wave * D))[lane];
  float s = v.x * v.x + v.y * v.y;
#pragma unroll
  for (int m = 16; m; m >>= 1) s += __shfl_xor(s, m, 32);
  const float scale = 1.0f / fmaxf(sqrtf(s), EPS);
  float2 o; o.x = v.x * scale; o.y = v.y * scale;
  ((float2*)(y + (size_t)wave * D))[lane] = o;
}

// ---------------------------------------------------------------------------
// 1b) Transpose V: [BH, N, D] -> Vt [BH, D, N]. 32x32 LDS tiles (+1 pad to
//     dodge bank conflicts); makes the AV GEMM's B operand k-contiguous so it
//     streams as b128 like the A operand.
// ---------------------------------------------------------------------------
__global__ void fns_transpose_v(const float* __restrict__ v,
                                float* __restrict__ vt) {
  using namespace fns;
  __shared__ float tile[32][33];
  const int bh = blockIdx.z;
  const int k0 = blockIdx.x * 32;
  const int d0 = blockIdx.y * 32;
  const int tx = threadIdx.x & 31;
  const int ty = threadIdx.x >> 5;            // 8 rows per pass
  const float* vb  = v  + (size_t)bh * N * D;
  float*       vtb = vt + (size_t)bh * D * N;
#pragma unroll
  for (int r = 0; r < 32; r += 8)
    tile[ty + r][tx] = vb[(size_t)(k0 + ty + r) * D + (d0 + tx)];
  __syncthreads();
#pragma unroll
  for (int r = 0; r < 32; r += 8)
    vtb[(size_t)(d0 + ty + r) * N + (k0 + tx)] = tile[tx][ty + r];
}

// ---------------------------------------------------------------------------
// 2) Score tiles: S = (1 + arccos(clip(qn . kn^T)))^(-65.5), causal.
//    One wave per 16x16 tile. Masked tiles (tj > ti) exit immediately: the
//    rescale pass synthesizes the zeros. Each b128 load feeds two WMMA
//    k-steps (k-index remap is legal: dot products are order-free here).
//    Per-tile row/col partial sums are wave-reduced and written for a
//    deterministic two-level reduction (no float atomics).
// ---------------------------------------------------------------------------
__global__ void fns_score(const float* __restrict__ qn,
                          const float* __restrict__ kn,
                          float* __restrict__ S,
                          float* __restrict__ rowpart,
                          float* __restrict__ colpart) {
  using namespace fns;
  const int tilesPerHead = NT * NT;
  const int wave = (int)((blockIdx.x * blockDim.x + threadIdx.x) >> 5);
  const int lane = threadIdx.x & 31;
  if (wave >= BH * tilesPerHead) return;

  const int bh = wave / tilesPerHead;
  const int t  = wave - bh * tilesPerHead;
  const int ti = t / NT;
  const int tj = t - ti * NT;
  if (tj > ti) return;                        // fully masked tile -> skip

  const int i0 = ti * 16, j0 = tj * 16;
  const int ln   = lane & 15;                 // A row M | B col N | C/D col N
  const int half = lane >> 4;

  const float4* qrow4 = (const float4*)(qn + ((size_t)bh * N + i0 + ln) * D);
  const float4* krow4 = (const float4*)(kn + ((size_t)bh * N + j0 + ln) * D);

  v8f c = {};
#pragma unroll
  for (int s = 0; s < 8; ++s) {               // k group: half0 8s..+3, half1 8s+4..+7
    const float4 aq = qrow4[2 * s + half];
    const float4 bk = krow4[2 * s + half];
    v2f a1 = { aq.x, aq.y }, b1 = { bk.x, bk.y };
    c = WMMA_F32X4(a1, b1, c);
    v2f a2 = { aq.z, aq.w }, b2 = { bk.z, bk.w };
    c = WMMA_F32X4(a2, b2, c);
  }

  float* Sbh = S + (size_t)bh * N * N;
  float vr[8];
  float cp = 0.0f;
#pragma unroll
  for (int r = 0; r < 8; ++r) {
    const int ig = i0 + r + 8 * half;         // C/D: M = vgpr + 8*half
    const int jg = j0 + ln;                   // C/D: N = lane&15
    float val;
    if (jg > ig) {                            // diagonal-tile masked entries
      val = 0.0f;                             // (1+10000)^-65.5 underflows
    } else {
      const float cs = fminf(fmaxf(c[r], -1.0f), 1.0f);
      val = expf(EXPONENT * log1pf(acosf(cs)));   // (1+g)^-65.5, radius=1
    }
    Sbh[(size_t)ig * N + jg] = val;
    vr[r] = val;
    cp += val;
  }

  // Column partial: 16 rows of column (j0+ln) in this tile.
  cp += __shfl_xor(cp, 16, 32);
  if (lane < 16)
    colpart[((size_t)bh * N + j0 + ln) * NT + ti] = cp;

  // Row partials: 16 columns of each row in this tile (reduce within half).
#pragma unroll
  for (int r = 0; r < 8; ++r) {
    float rs = vr[r];
#pragma unroll
    for (int m = 8; m; m >>= 1) rs += __shfl_xor(rs, m, 32);
    if (ln == 0)
      rowpart[((size_t)bh * N + i0 + r + 8 * half) * NT + tj] = rs;
  }
}

// ---------------------------------------------------------------------------
// 3) Deterministic second-level reductions over tile partials.
// ---------------------------------------------------------------------------
__global__ void fns_reduce_rows(const float* __restrict__ rowpart,
                                float* __restrict__ n_r) {
  using namespace fns;
  const int row = (int)(blockIdx.x * blockDim.x + threadIdx.x);
  if (row >= BH * N) return;
  const int ti = (row & (N - 1)) >> 4;        // unmasked col tiles: 0..ti
  const float* p = rowpart + (size_t)row * NT;
  float s = 0.0f;
  for (int tj = 0; tj <= ti; ++tj) s += p[tj];
  n_r[row] = s;
}

__global__ void fns_reduce_cols(const float* __restrict__ colpart,
                                float* __restrict__ n_c) {
  using namespace fns;
  const int col = (int)(blockIdx.x * blockDim.x + threadIdx.x);
  if (col >= BH * N) return;
  const int tj = (col & (N - 1)) >> 4;        // unmasked row tiles: tj..NT-1
  const float* p = colpart + (size_t)col * NT;
  float s = 0.0f;
  for (int ti = tj; ti < NT; ++ti) s += p[ti];
  n_c[col] = s;
}

// ---------------------------------------------------------------------------
// 4) Fused t + rescale, one wave per row, row cached in 16 float4 registers.
//    W = (nr^-1/2 * S * nc^-1/2) / max(nr^-1/2 * t, EPS),
//    t = sum_j S * nc^-1/2. Masked cols (j > i) synthesized as zero (their S
//    was never written). In-place in the attn_weights output region.
// ---------------------------------------------------------------------------
__global__ void fns_rescale(float* __restrict__ S,
                            const float* __restrict__ n_r,
                            const float* __restrict__ n_c) {
  using namespace fns;
  const int wave = (int)((blockIdx.x * blockDim.x + threadIdx.x) >> 5);
  const int lane = threadIdx.x & 31;
  if (wave >= BH * N) return;
  const int bh = wave >> 11;                  // / N
  const int ih = wave & (N - 1);              // row index within head

  float4* row = (float4*)(S + (size_t)wave * N);
  const float4* ncb = (const float4*)(n_c + (size_t)bh * N);

  float4 buf[16];
  float ts = 0.0f;
#pragma unroll
  for (int it = 0; it < 16; ++it) {
    const int j4 = lane + 32 * it;
    const int jb = 4 * j4;
    float4 p = { 0.0f, 0.0f, 0.0f, 0.0f };
    if (jb <= ih) {                           // at least one unmasked element
      const float4 v  = row[j4];
      const float4 nc = ncb[j4];
      p.x = (jb + 0 <= ih) ? v.x * rsqrtf(nc.x) : 0.0f;
      p.y = (jb + 1 <= ih) ? v.y * rsqrtf(nc.y) : 0.0f;
      p.z = (jb + 2 <= ih) ? v.z * rsqrtf(nc.z) : 0.0f;
      p.w = (jb + 3 <= ih) ? v.w * rsqrtf(nc.w) : 0.0f;
      ts += (p.x + p.y) + (p.z + p.w);
    }
    buf[it] = p;
  }
#pragma unroll
  for (int m = 16; m; m >>= 1) ts += __shfl_xor(ts, m, 32);

  const float rn    = rsqrtf(n_r[wave]);
  const float denom = fmaxf(rn * ts, EPS);
  const float scale = rn / denom;
#pragma unroll
  for (int it = 0; it < 16; ++it) {
    float4 p = buf[it];
    p.x *= scale; p.y *= scale; p.z *= scale; p.w *= scale;
    row[lane + 32 * it] = p;
  }
}

// ---------------------------------------------------------------------------
// 5) output = W @ v using Vt [BH, D, N]. One wave per (bh, 16-row tile):
//    16x64 output tile in four v8f accumulators; causal K truncation.
//    Both operands now stream as b128: one float4 of W and one float4 per
//    column-tile of Vt feed two WMMA k-steps each.
// ---------------------------------------------------------------------------
__global__ void fns_av(const float* __restrict__ W, const float* __restrict__ vt,
                       float* __restrict__ out) {
  using namespace fns;
  const int wave = (int)((blockIdx.x * blockDim.x + threadIdx.x) >> 5);
  const int lane = threadIdx.x & 31;
  if (wave >= BH * NT) return;
  const int bh = wave / NT;
  const int ti = wave - bh * NT;
  const int ln   = lane & 15;
  const int half = lane >> 4;

  const float4* Wrow4 = (const float4*)(W + ((size_t)bh * N + ti * 16 + ln) * N);
  const float* Vt0 = vt + ((size_t)bh * D + 0 * 16 + ln) * N;   // column ln of
  const float* Vt1 = vt + ((size_t)bh * D + 1 * 16 + ln) * N;   // each 16-wide
  const float* Vt2 = vt + ((size_t)bh * D + 2 * 16 + ln) * N;   // d tile
  const float* Vt3 = vt + ((size_t)bh * D + 3 * 16 + ln) * N;

  v8f acc0 = {}, acc1 = {}, acc2 = {}, acc3 = {};
  const int ngroups = ((ti + 1) * 16) >> 3;   // causal: k < (ti+1)*16
  for (int s = 0; s < ngroups; ++s) {
    const int g = 2 * s + half;               // float4 index: k = 4g .. 4g+3
    const float4 aw = Wrow4[g];
    v2f a1 = { aw.x, aw.y };
    v2f a2 = { aw.z, aw.w };
    const float4 b0 = ((const float4*)Vt0)[g];
    const float4 b1 = ((const float4*)Vt1)[g];
    const float4 b2 = ((const float4*)Vt2)[g];
    const float4 b3 = ((const float4*)Vt3)[g];
    { v2f bb = { b0.x, b0.y }; acc0 = WMMA_F32X4(a1, bb, acc0); }
    { v2f bb = { b0.z, b0.w }; acc0 = WMMA_F32X4(a2, bb, acc0); }
    { v2f bb = { b1.x, b1.y }; acc1 = WMMA_F32X4(a1, bb, acc1); }
    { v2f bb = { b1.z, b1.w }; acc1 = WMMA_F32X4(a2, bb, acc1); }
    { v2f bb = { b2.x, b2.y }; acc2 = WMMA_F32X4(a1, bb, acc2); }
    { v2f bb = { b2.z, b2.w }; acc2 = WMMA_F32X4(a2, bb, acc2); }
    { v2f bb = { b3.x, b3.y }; acc3 = WMMA_F32X4(a1, bb, acc3); }
    { v2f bb = { b3.z, b3.w }; acc3 = WMMA_F32X4(a2, bb, acc3); }
  }

  float* Ob = out + ((size_t)bh * N + ti * 16) * D;
#pragma unroll
  for (int r = 0; r < 8; ++r) {
    const size_t ro = (size_t)(r + 8 * half) * D + ln;
    Ob[ro + 0]  = acc0[r];
    Ob[ro + 16] = acc1[r];
    Ob[ro + 32] = acc2[r];
    Ob[ro + 48] = acc3[r];
  }
}

// ---------------------------------------------------------------------------
// Launch
// ---------------------------------------------------------------------------
extern "C" void kernel_launch(void* const* d_in, const int* in_sizes, int n_in,
                              void* d_out, int out_size, void* d_ws, size_t ws_size,
                              hipStream_t stream) {
  using namespace fns;
  (void)in_sizes; (void)n_in; (void)out_size; (void)ws_size;

  const float* q = (const float*)d_in[0];
  const float* k = (const float*)d_in[1];
  const float* v = (const float*)d_in[2];
  // d_in[3] = attn_mask (causal triu, k=1): structure known, not read.

  // Output layout: [output B,H,N,D][attn_weights B,H,N,N]
  float* out_o = (float*)d_out;
  float* out_w = out_o + (size_t)BH * N * D;

  // Workspace: qn | kn | rowpart | colpart | n_r | n_c | vt   (~89 MB)
  float* qn      = (float*)d_ws;
  float* kn      = qn      + (size_t)BH * N * D;
  float* rowpart = kn      + (size_t)BH * N * D;
  float* colpart = rowpart + (size_t)BH * N * NT;
  float* n_r     = colpart + (size_t)BH * N * NT;
  float* n_c     = n_r     + (size_t)BH * N;
  float* vt      = n_c     + (size_t)BH * N;

  const int rows = BH * N;                          // 49152

  // 1) normalize q, k (one wave per row); transpose V (independent)
  fns_l2norm<<<(rows * 32 + 255) / 256, 256, 0, stream>>>(q, qn, rows);
  fns_l2norm<<<(rows * 32 + 255) / 256, 256, 0, stream>>>(k, kn, rows);
  fns_transpose_v<<<dim3(N / 32, D / 32, BH), 256, 0, stream>>>(v, vt);

  // 2) scores + tile partial sums (one wave per 16x16 tile; masked tiles exit)
  {
    const int waves = BH * NT * NT;                 // 393216
    fns_score<<<(waves * 32 + 255) / 256, 256, 0, stream>>>(qn, kn, out_w,
                                                            rowpart, colpart);
  }

  // 3) deterministic reductions -> n_r, n_c
  fns_reduce_rows<<<(rows + 255) / 256, 256, 0, stream>>>(rowpart, n_r);
  fns_reduce_cols<<<(rows + 255) / 256, 256, 0, stream>>>(colpart, n_c);

  // 4) fused t + rescale, in place -> attn_weights (one wave per row)
  fns_rescale<<<(rows * 32 + 255) / 256, 256, 0, stream>>>(out_w, n_r, n_c);

  // 5) output = W @ Vt (one wave per 16-row tile)
  {
    const int waves = BH * NT;                      // 3072
    fns_av<<<(waves * 32 + 255) / 256, 256, 0, stream>>>(out_w, vt, out_o);
  }
}